// VariationalSNLDS_9028021256618
// MI455X (gfx1250) — compile-verified
//
#include <hip/hip_runtime.h>
#include <hip/hip_bf16.h>
#include <math.h>

// ---------------------------------------------------------------------------
// VariationalSNLDS forward for MI455X (gfx1250), fp32 end-to-end.
// All dense GEMMs use V_WMMA_F32_16X16X4_F32 (native fp32 WMMA on CDNA5).
// Problem is bandwidth/latency bound (~9 GFLOP vs ~40MB @ 23.3 TB/s), so
// fp32 WMMA is the right precision: matches reference numerics, compute is
// nowhere near the matrix ceiling anyway.
// ---------------------------------------------------------------------------

typedef __attribute__((ext_vector_type(2))) float v2f;
typedef __attribute__((ext_vector_type(8))) float v8f;

#define BDIM 64
#define TDIM 512
#define DOBS 128
#define DLAT 16
#define DHID 256
#define KMIX 8

#define LOG2PI 1.8378770664093453f

// workspace layout (float offsets)
#define WS_ACC      0                 // [0]=recon_sum [1]=logq_sum [2]=logZ_sum
#define WS_LOGPI    4                 // 8
#define WS_QLOGT    12                // 64: Qlog[k][j] = log_softmax(Q,-1)[j][k]
#define WS_LOGDET0  76                // 8
#define WS_LOGDETT  84                // 8
#define WS_L0       96                // 8*256
#define WS_LT       (96 + 2048)       // 8*256
#define WS_EV       (96 + 4096)       // B*T*K = 262144
// total: 266336 floats ~= 1.04 MB

// ---------------- WMMA helpers (V_WMMA_F32_16X16X4_F32) --------------------
// A 16x4 (MxK): lane m = lane&15; VGPR0 = K=(lane<16?0:2)+kbase, VGPR1 = +1
// B 4x16 (KxN): lane n = lane&15; VGPR0 = K=(lane<16?0:2)+kbase, VGPR1 = +1
// C/D 16x16:    lane n = lane&15; VGPR v -> M = v + 8*(lane>=16)

__device__ __forceinline__ v8f wmma4(v2f a, v2f b, v8f c) {
  return __builtin_amdgcn_wmma_f32_16x16x4_f32(
      /*neg_a=*/false, a, /*neg_b=*/false, b,
      /*c_mod=*/(short)0, c, /*reuse_a=*/false, /*reuse_b=*/false);
}

__device__ __forceinline__ v2f frag_a(const float* p, int ld, int kbase, int lane) {
  int m  = lane & 15;
  int kk = kbase + ((lane & 16) ? 2 : 0);
  const float* q = p + m * ld + kk;
  v2f a; a[0] = q[0]; a[1] = q[1];
  return a;
}

__device__ __forceinline__ v2f frag_b(const float* p, int ld, int kbase, int lane) {
  int n  = lane & 15;
  int kk = kbase + ((lane & 16) ? 2 : 0);
  const float* q = p + kk * ld + n;
  v2f b; b[0] = q[0]; b[1] = q[ld];
  return b;
}

// forward substitution: L y = diff (L 16x16 row-major lower), return |y|^2
__device__ __forceinline__ float solve_quad(const float* __restrict__ L,
                                            const float* __restrict__ diff) {
  float y[16]; float ss = 0.f;
  for (int i = 0; i < 16; i++) {
    float s = diff[i];
    for (int j = 0; j < i; j++) s -= L[i*16 + j] * y[j];
    float yi = s / L[i*16 + i];
    y[i] = yi; ss += yi * yi;
  }
  return ss;
}

// ---------------------------------------------------------------------------
__global__ void k_init(float* ws) {
  if (threadIdx.x < 4) ws[WS_ACC + threadIdx.x] = 0.f;
}

// Cholesky of C C^T + 1e-6 I for init_cov and covs; log_pi; Qlog^T.
__global__ void k_prep(const float* __restrict__ init_cov,
                       const float* __restrict__ covs,
                       const float* __restrict__ Q,
                       const float* __restrict__ pi,
                       float* __restrict__ ws) {
  int tid = threadIdx.x;
  if (tid < 16) {
    int k = tid & 7;
    const float* C = (tid < 8 ? init_cov : covs) + k * 256;
    float* Ldst = ws + (tid < 8 ? WS_L0 : WS_LT) + k * 256;
    float* ldd  = ws + (tid < 8 ? WS_LOGDET0 : WS_LOGDETT) + k;
    float M[16][16];
    for (int i = 0; i < 16; i++)
      for (int j = 0; j < 16; j++) {
        float s = (i == j) ? 1e-6f : 0.f;
        for (int p = 0; p < 16; p++) s += C[i*16 + p] * C[j*16 + p];
        M[i][j] = s;
      }
    float logdet = 0.f;
    for (int j = 0; j < 16; j++) {
      float s = M[j][j];
      for (int p = 0; p < j; p++) s -= M[j][p] * M[j][p];
      float d = sqrtf(s);
      M[j][j] = d;
      logdet += 2.f * logf(d);
      for (int i = j + 1; i < 16; i++) {
        float t = M[i][j];
        for (int p = 0; p < j; p++) t -= M[i][p] * M[j][p];
        M[i][j] = t / d;
      }
    }
    for (int i = 0; i < 16; i++)
      for (int j = 0; j < 16; j++)
        Ldst[i*16 + j] = (j <= i) ? M[i][j] : 0.f;
    *ldd = logdet;
  } else if (tid == 16) {
    float m = -1e30f;
    for (int i = 0; i < 8; i++) m = fmaxf(m, pi[i]);
    float s = 0.f;
    for (int i = 0; i < 8; i++) s += expf(pi[i] - m);
    float lse = m + logf(s);
    for (int i = 0; i < 8; i++) ws[WS_LOGPI + i] = pi[i] - lse;
  } else if (tid == 17) {
    for (int j = 0; j < 8; j++) {
      float m = -1e30f;
      for (int i = 0; i < 8; i++) m = fmaxf(m, Q[j*8 + i]);
      float s = 0.f;
      for (int i = 0; i < 8; i++) s += expf(Q[j*8 + i] - m);
      float lse = m + logf(s);
      for (int k = 0; k < 8; k++) ws[WS_QLOGT + k*8 + j] = Q[j*8 + k] - lse;
    }
  }
}

// Encoder: h = leaky(x@W1+b1); enc = h@W2+b2; z = mu + exp(.5 lv) eps; logq.
__global__ __launch_bounds__(256) void k_encoder(
    const float* __restrict__ x, const float* __restrict__ eps,
    const float* __restrict__ W1, const float* __restrict__ b1,
    const float* __restrict__ W2, const float* __restrict__ b2,
    float* __restrict__ z_out, float* __restrict__ ws) {
  __shared__ float Xs[16][DOBS + 4];
  __shared__ float Hs[16][DHID + 4];
  __shared__ float Es[16][2 * DLAT];
  __shared__ float red[256];
  const int tid = threadIdx.x, lane = tid & 31, wv = tid >> 5;
  const int R = blockIdx.x * 16;   // rows = flattened (b,t)

  for (int i = tid; i < 16 * DOBS; i += 256)
    Xs[i >> 7][i & 127] = x[(R + (i >> 7)) * DOBS + (i & 127)];
  __syncthreads();

  for (int half = 0; half < 2; half++) {        // wave -> 2 n-tiles of h
    int nt = 2 * wv + half;
    v8f c = {};
    for (int ks = 0; ks < DOBS / 4; ks++) {
      v2f a = frag_a(&Xs[0][0], DOBS + 4, 4 * ks, lane);
      v2f b = frag_b(W1 + nt * 16, DHID, 4 * ks, lane);
      c = wmma4(a, b, c);
    }
    float bias = b1[nt * 16 + (lane & 15)];
    int mb = (lane & 16) ? 8 : 0;
    for (int v = 0; v < 8; v++) {
      float h = c[v] + bias;
      Hs[mb + v][nt * 16 + (lane & 15)] = h > 0.f ? h : 0.01f * h;
    }
  }
  __syncthreads();

  if (wv < 2) {                                  // enc: 32 cols
    int nt = wv;
    v8f c = {};
    for (int ks = 0; ks < DHID / 4; ks++) {
      v2f a = frag_a(&Hs[0][0], DHID + 4, 4 * ks, lane);
      v2f b = frag_b(W2 + nt * 16, 2 * DLAT, 4 * ks, lane);
      c = wmma4(a, b, c);
    }
    float bias = b2[nt * 16 + (lane & 15)];
    int mb = (lane & 16) ? 8 : 0;
    for (int v = 0; v < 8; v++)
      Es[mb + v][nt * 16 + (lane & 15)] = c[v] + bias;
  }
  __syncthreads();

  {
    int m = tid >> 4, cix = tid & 15;
    int row = R + m;
    float mu = Es[m][cix], lv = Es[m][DLAT + cix];
    float e = eps[row * DLAT + cix];
    z_out[row * DLAT + cix] = mu + expf(0.5f * lv) * e;
    // logq element: -(z-mu)^2/exp(lv) == -e^2
    red[tid] = -0.5f * (LOG2PI + lv + e * e);
  }
  __syncthreads();
  for (int s = 128; s > 0; s >>= 1) {
    if (tid < s) red[tid] += red[tid + s];
    __syncthreads();
  }
  if (tid == 0) atomicAdd(&ws[WS_ACC + 1], red[0]);
}

// Decoder: x_hat = leaky(z@Wd1+bd1)@Wd2+bd2; recon accumulation.
__global__ __launch_bounds__(256) void k_decoder(
    const float* __restrict__ x, const float* __restrict__ z,
    const float* __restrict__ Wd1, const float* __restrict__ bd1,
    const float* __restrict__ Wd2, const float* __restrict__ bd2,
    float* __restrict__ xhat, float* __restrict__ ws) {
  __shared__ float Zs[16][20];
  __shared__ float Hs[16][DHID + 4];
  __shared__ float red[256];
  const int tid = threadIdx.x, lane = tid & 31, wv = tid >> 5;
  const int R = blockIdx.x * 16;

  Zs[tid >> 4][tid & 15] = z[(R + (tid >> 4)) * DLAT + (tid & 15)];
  __syncthreads();

  for (int half = 0; half < 2; half++) {
    int nt = 2 * wv + half;
    v8f c = {};
    for (int ks = 0; ks < 4; ks++) {
      v2f a = frag_a(&Zs[0][0], 20, 4 * ks, lane);
      v2f b = frag_b(Wd1 + nt * 16, DHID, 4 * ks, lane);
      c = wmma4(a, b, c);
    }
    float bias = bd1[nt * 16 + (lane & 15)];
    int mb = (lane & 16) ? 8 : 0;
    for (int v = 0; v < 8; v++) {
      float h = c[v] + bias;
      Hs[mb + v][nt * 16 + (lane & 15)] = h > 0.f ? h : 0.01f * h;
    }
  }
  __syncthreads();

  float rl = 0.f;
  {
    int nt = wv;                                 // 8 waves x 16 cols = 128
    v8f c = {};
    for (int ks = 0; ks < DHID / 4; ks++) {
      v2f a = frag_a(&Hs[0][0], DHID + 4, 4 * ks, lane);
      v2f b = frag_b(Wd2 + nt * 16, DOBS, 4 * ks, lane);
      c = wmma4(a, b, c);
    }
    int n = nt * 16 + (lane & 15);
    float bias = bd2[n];
    int mb = (lane & 16) ? 8 : 0;
    for (int v = 0; v < 8; v++) {
      int row = R + mb + v;
      float xh = c[v] + bias;
      xhat[row * DOBS + n] = xh;
      float d = x[row * DOBS + n] - xh;
      // -0.5*log(2*pi*VAR) - d^2/(2*VAR), VAR=5e-4
      rl += 2.8815127f - d * d * 1000.0f;
    }
  }
  red[tid] = rl; __syncthreads();
  for (int s = 128; s > 0; s >>= 1) {
    if (tid < s) red[tid] += red[tid + s];
    __syncthreads();
  }
  if (tid == 0) atomicAdd(&ws[WS_ACC + 0], red[0]);
}

// Transition: ht = softplus(z@Wt1[k]+bt1[k]); means = ht@Wt2[k]+bt2[k];
// ev[b,t+1,k] = mvn_logprob(z[b,t+1]-means, Lt[k]).
__global__ __launch_bounds__(256) void k_trans(
    const float* __restrict__ z, const float* __restrict__ Wt1,
    const float* __restrict__ bt1, const float* __restrict__ Wt2,
    const float* __restrict__ bt2, float* __restrict__ ws) {
  __shared__ float Zs[16][20];
  __shared__ float Hs[16][DHID + 4];
  __shared__ float Ps[8][16][16];
  const int tid = threadIdx.x, lane = tid & 31, wv = tid >> 5;
  const int tt = blockIdx.x, k = blockIdx.y, b = blockIdx.z;
  const int t0 = tt * 16;

  Zs[tid >> 4][tid & 15] = z[(b * TDIM + t0 + (tid >> 4)) * DLAT + (tid & 15)];
  __syncthreads();

  const float* W1k = Wt1 + k * DLAT * DHID;      // (16,256) row-major
  for (int half = 0; half < 2; half++) {
    int nt = 2 * wv + half;
    v8f c = {};
    for (int ks = 0; ks < 4; ks++) {
      v2f a  = frag_a(&Zs[0][0], 20, 4 * ks, lane);
      v2f bf = frag_b(W1k + nt * 16, DHID, 4 * ks, lane);
      c = wmma4(a, bf, c);
    }
    float bias = bt1[k * DHID + nt * 16 + (lane & 15)];
    int mb = (lane & 16) ? 8 : 0;
    for (int v = 0; v < 8; v++) {
      float hx = c[v] + bias;
      Hs[mb + v][nt * 16 + (lane & 15)] =
          (hx > 20.f) ? hx : log1pf(expf(hx));   // softplus
    }
  }
  __syncthreads();

  {
    const float* W2k = Wt2 + k * DHID * DLAT;    // (256,16) row-major
    v8f c = {};
    for (int ks = 8 * wv; ks < 8 * wv + 8; ks++) {   // K split across 8 waves
      v2f a  = frag_a(&Hs[0][0], DHID + 4, 4 * ks, lane);
      v2f bf = frag_b(W2k, DLAT, 4 * ks, lane);
      c = wmma4(a, bf, c);
    }
    int n = lane & 15, mb = (lane & 16) ? 8 : 0;
    for (int v = 0; v < 8; v++) Ps[wv][mb + v][n] = c[v];
  }
  __syncthreads();

  if (tid < 16) {
    int m = tid;
    int t_tgt = t0 + m + 1;
    if (t_tgt < TDIM) {
      float mean[16];
      for (int n = 0; n < 16; n++) {
        float s = bt2[k * DLAT + n];
        for (int w = 0; w < 8; w++) s += Ps[w][m][n];
        mean[n] = s;
      }
      const float* zt = z + (b * TDIM + t_tgt) * DLAT;
      float diff[16];
      for (int n = 0; n < 16; n++) diff[n] = zt[n] - mean[n];
      float ss = solve_quad(ws + WS_LT + k * 256, diff);
      ws[WS_EV + ((b * TDIM + t_tgt) * KMIX + k)] =
          -0.5f * (16.f * LOG2PI + ws[WS_LOGDETT + k] + ss);
    }
  }
}

// ev at t=0 against init_mean / L0.
__global__ void k_ev1(const float* __restrict__ z,
                      const float* __restrict__ init_mean,
                      float* __restrict__ ws) {
  int tid = blockIdx.x * blockDim.x + threadIdx.x;
  if (tid >= BDIM * KMIX) return;
  int b = tid >> 3, k = tid & 7;
  const float* z0 = z + (size_t)(b * TDIM) * DLAT;
  float diff[16];
  for (int i = 0; i < 16; i++) diff[i] = z0[i] - init_mean[k * DLAT + i];
  float ss = solve_quad(ws + WS_L0 + k * 256, diff);
  ws[WS_EV + (b * TDIM) * KMIX + k] =
      -0.5f * (16.f * LOG2PI + ws[WS_LOGDET0 + k] + ss);
}

// HMM forward scan in log space: one wave per batch, sequential over T.
__global__ void k_scan(float* __restrict__ ws) {
  __shared__ float QT[64], lpi[8], la[8], lp[8];
  __shared__ float lZs;
  const int tid = threadIdx.x;
  const int b = blockIdx.x;
  for (int i = tid; i < 64; i += 32) QT[i] = ws[WS_QLOGT + i];
  if (tid < 8) lpi[tid] = ws[WS_LOGPI + tid];
  __syncthreads();

  const float* ev = ws + WS_EV + (size_t)b * TDIM * KMIX;
  float acc = 0.f;

  if (tid < 8) lp[tid] = ev[tid] + lpi[tid];
  __syncthreads();
  if (tid == 0) {
    float m = -1e30f;
    for (int i = 0; i < 8; i++) m = fmaxf(m, lp[i]);
    float s = 0.f;
    for (int i = 0; i < 8; i++) s += expf(lp[i] - m);
    lZs = m + logf(s);
  }
  __syncthreads();
  if (tid < 8) la[tid] = lp[tid] - lZs;
  if (tid == 0) acc = lZs;
  __syncthreads();

  for (int t = 1; t < TDIM; t++) {
    if (tid < 8) {
      float e = ev[t * KMIX + tid];
      float m = -1e30f, v[8];
      for (int j = 0; j < 8; j++) { v[j] = QT[tid * 8 + j] + la[j]; m = fmaxf(m, v[j]); }
      float s = 0.f;
      for (int j = 0; j < 8; j++) s += expf(v[j] - m);
      lp[tid] = e + m + logf(s);
    }
    __syncthreads();
    if (tid == 0) {
      float m = -1e30f;
      for (int i = 0; i < 8; i++) m = fmaxf(m, lp[i]);
      float s = 0.f;
      for (int i = 0; i < 8; i++) s += expf(lp[i] - m);
      float lz = m + logf(s);
      lZs = lz; acc += lz;
    }
    __syncthreads();
    if (tid < 8) la[tid] = lp[tid] - lZs;
    __syncthreads();
  }
  if (tid == 0) atomicAdd(&ws[WS_ACC + 2], acc);
}

__global__ void k_final(const float* __restrict__ ws, float* __restrict__ loss) {
  float recon = ws[WS_ACC + 0] / (float)BDIM;
  float logq  = ws[WS_ACC + 1] / (float)BDIM;   // sum(logq)/B
  float msm   = ws[WS_ACC + 2] / (float)BDIM;
  *loss = -(recon - logq + msm);                // entropy_q = -sum(logq)/B
}

// ---------------------------------------------------------------------------
extern "C" void kernel_launch(void* const* d_in, const int* in_sizes, int n_in,
                              void* d_out, int out_size, void* d_ws, size_t ws_size,
                              hipStream_t stream) {
  const float* x   = (const float*)d_in[0];
  const float* eps = (const float*)d_in[1];
  const float* W1  = (const float*)d_in[2];
  const float* b1  = (const float*)d_in[3];
  const float* W2  = (const float*)d_in[4];
  const float* b2  = (const float*)d_in[5];
  const float* Wt1 = (const float*)d_in[6];
  const float* bt1 = (const float*)d_in[7];
  const float* Wt2 = (const float*)d_in[8];
  const float* bt2 = (const float*)d_in[9];
  const float* Wd1 = (const float*)d_in[10];
  const float* bd1 = (const float*)d_in[11];
  const float* Wd2 = (const float*)d_in[12];
  const float* bd2 = (const float*)d_in[13];
  const float* Q   = (const float*)d_in[14];
  const float* pi  = (const float*)d_in[15];
  const float* im  = (const float*)d_in[16];
  const float* ic  = (const float*)d_in[17];
  const float* cv  = (const float*)d_in[18];

  float* out  = (float*)d_out;
  float* xhat = out;
  float* z    = out + (size_t)BDIM * TDIM * DOBS;
  float* loss = z + (size_t)BDIM * TDIM * DLAT;
  float* ws   = (float*)d_ws;

  k_init<<<1, 32, 0, stream>>>(ws);
  k_prep<<<1, 64, 0, stream>>>(ic, cv, Q, pi, ws);
  k_encoder<<<BDIM * TDIM / 16, 256, 0, stream>>>(x, eps, W1, b1, W2, b2, z, ws);
  k_decoder<<<BDIM * TDIM / 16, 256, 0, stream>>>(x, z, Wd1, bd1, Wd2, bd2, xhat, ws);
  k_ev1<<<1, BDIM * KMIX, 0, stream>>>(z, im, ws);
  k_trans<<<dim3(TDIM / 16, KMIX, BDIM), 256, 0, stream>>>(z, Wt1, bt1, Wt2, bt2, ws);
  k_scan<<<BDIM, 32, 0, stream>>>(ws);
  k_final<<<1, 1, 0, stream>>>(ws, loss);
}